// WarpController_7164005450020
// MI455X (gfx1250) — compile-verified
//
#include <hip/hip_runtime.h>

#define F_DIM 8192
#define HID   128
#define B_N   32
#define K_N   16
#define HH    256
#define WW    256
#define HW    65536

typedef float v2f __attribute__((ext_vector_type(2)));
typedef float v8f __attribute__((ext_vector_type(8)));

// --------------------------------------------------------------------------
// Kernel 1: h_raw = x @ W1   (32x8192 @ 8192x128, fp32) via V_WMMA_F32_16X16X4_F32
// One wave (32 threads) per 16x16 output tile; 2x8 tiles -> 16 blocks.
// A layout (ISA 7.12.2, 32-bit A 16x4): lanes 0-15 K={0,1}, lanes 16-31 K={2,3}.
// C/D layout: VGPR r -> row (r + 8*half), col = lane&15.
// --------------------------------------------------------------------------
__global__ __launch_bounds__(32) void k_gemm_h(const float* __restrict__ x,
                                               const float* __restrict__ W1,
                                               float* __restrict__ h_raw) {
  const int tn   = blockIdx.x & 7;   // 8 N-tiles over 128
  const int tm   = blockIdx.x >> 3;  // 2 M-tiles over 32
  const int lane = threadIdx.x;
  const int half = lane >> 4;        // 0: K pair {0,1}; 1: K pair {2,3}
  const int l    = lane & 15;
  const int arow = tm * 16 + l;
  const int bcol = tn * 16 + l;
  const float* ap = x + (size_t)arow * F_DIM + 2 * half;

  v8f acc = {};
#pragma unroll 8
  for (int k = 0; k < F_DIM; k += 4) {
    v2f a, b;
    a.x = ap[k];
    a.y = ap[k + 1];
    const float* bp = W1 + (size_t)(k + 2 * half) * HID + bcol;
    b.x = bp[0];
    b.y = bp[HID];
    // (neg_a, A, neg_b, B, c_mod, C, reuse_a, reuse_b)
    acc = __builtin_amdgcn_wmma_f32_16x16x4_f32(false, a, false, b, (short)0, acc,
                                                false, false);
  }
#pragma unroll
  for (int r = 0; r < 8; ++r) {
    const int row = tm * 16 + half * 8 + r;
    h_raw[row * HID + bcol] = acc[r];
  }
}

// --------------------------------------------------------------------------
// Kernel 2: LayerNorm + ReLU + ldmark heads. One block (128 threads) per batch.
// pred = relu(ln(h)) @ W2 + ldmark_mean ; diff = scales * (relu(ln(h)) @ W3)
// --------------------------------------------------------------------------
__global__ __launch_bounds__(128) void k_ln_ldmark(
    const float* __restrict__ h_raw, const float* __restrict__ b1,
    const float* __restrict__ gamma, const float* __restrict__ beta,
    const float* __restrict__ W2, const float* __restrict__ W3,
    const float* __restrict__ ldmark_mean, const float* __restrict__ scales,
    float* __restrict__ pred_out, float* __restrict__ diff_out) {
  const int b = blockIdx.x, t = threadIdx.x;
  __shared__ float s1[128], s2[128], sh[128];
  float v = h_raw[b * HID + t] + b1[t];
  s1[t] = v;
  s2[t] = v * v;
  __syncthreads();
  for (int s = 64; s > 0; s >>= 1) {
    if (t < s) { s1[t] += s1[t + s]; s2[t] += s2[t + s]; }
    __syncthreads();
  }
  const float mu  = s1[0] * (1.0f / 128.0f);
  const float var = s2[0] * (1.0f / 128.0f) - mu * mu;
  float hn = (v - mu) * rsqrtf(var + 1e-5f) * gamma[t] + beta[t];
  hn = fmaxf(hn, 0.0f);
  sh[t] = hn;
  __syncthreads();
  if (t < 64) {
    const int j = t & 31;
    const float* M = (t < 32) ? W2 : W3;
    float dot = 0.0f;
#pragma unroll 8
    for (int i = 0; i < HID; ++i) dot = fmaf(sh[i], M[i * 32 + j], dot);
    if (t < 32) pred_out[b * 32 + j] = dot + ldmark_mean[j];
    else        diff_out[b * 32 + j] = scales[b] * dot;
  }
}

// --------------------------------------------------------------------------
// Kernel 3: per-batch 19x19 TPS solve, lane-parallel Gauss-Jordan with partial
// pivoting inside one wave32. Lane j<19 owns matrix column j; lanes 19,20 own
// the two RHS columns. Pivot column broadcast via __shfl (ds_bpermute).
// --------------------------------------------------------------------------
__global__ __launch_bounds__(32) void k_spline_solve(
    const float* __restrict__ pred, const float* __restrict__ diff,
    float* __restrict__ wv) {
  const int b = blockIdx.x;
  const int lane = threadIdx.x;

  float cy[K_N], cx[K_N];
#pragma unroll
  for (int i = 0; i < K_N; ++i) {
    cy[i] = pred[b * 32 + 2 * i]     + diff[b * 32 + 2 * i];
    cx[i] = pred[b * 32 + 2 * i + 1] + diff[b * 32 + 2 * i + 1];
  }

  float col[19];
#pragma unroll
  for (int i = 0; i < 19; ++i) col[i] = 0.0f;

  if (lane < K_N) {                  // columns 0..15: A = phi + reg; rows 16-18 = Bm^T
#pragma unroll
    for (int i = 0; i < K_N; ++i) {
      const float dy = cy[i] - cy[lane], dx = cx[i] - cx[lane];
      const float r2 = dy * dy + dx * dx;
      const float ph = 0.5f * r2 * __logf(fmaxf(r2, 1e-10f));
      col[i] = ph + ((i == lane) ? 1e-6f : 0.0f);
    }
    col[16] = cy[lane];
    col[17] = cx[lane];
    col[18] = 1.0f;
  } else if (lane == 16) {           // column 16: c[:,0]
    for (int i = 0; i < K_N; ++i) col[i] = cy[i];
  } else if (lane == 17) {           // column 17: c[:,1]
    for (int i = 0; i < K_N; ++i) col[i] = cx[i];
  } else if (lane == 18) {           // column 18: ones
    for (int i = 0; i < K_N; ++i) col[i] = 1.0f;
  } else if (lane == 19 || lane == 20) {  // RHS: f = dst - src = diff
    const int d = lane - 19;
    for (int i = 0; i < K_N; ++i) col[i] = diff[b * 32 + 2 * i + d];
  }

  for (int p = 0; p < 19; ++p) {
    float colp[19];
#pragma unroll
    for (int i = 0; i < 19; ++i) colp[i] = __shfl(col[i], p, 32);
    // partial pivot (identical decision on every lane)
    int piv = p;
    float best = fabsf(colp[p]);
    for (int i = p + 1; i < 19; ++i) {
      const float a = fabsf(colp[i]);
      if (a > best) { best = a; piv = i; }
    }
    float tp = col[p];  col[p]  = col[piv];  col[piv]  = tp;
    tp       = colp[p]; colp[p] = colp[piv]; colp[piv] = tp;
    const float inv = 1.0f / colp[p];
#pragma unroll
    for (int i = 0; i < 19; ++i) {
      if (i == p) continue;
      col[i] = fmaf(-colp[i] * inv, col[p], col[i]);
    }
  }
  // matrix now diagonal: x_i = rhs_i / M_ii
  float sol[19];
#pragma unroll
  for (int i = 0; i < 19; ++i) {
    const float dia = __shfl(col[i], i, 32);
    sol[i] = col[i] / dia;
  }
  if (lane == 19 || lane == 20) {
    const int d = lane - 19;
    for (int i = 0; i < 19; ++i) wv[b * 38 + 2 * i + d] = sol[i];
  }
}

// --------------------------------------------------------------------------
// Kernel 4: fused TPS flow evaluation + bilinear warp. 256 blocks per batch,
// each block = 256 consecutive pixels (one image row). Coefficients in LDS.
// --------------------------------------------------------------------------
__global__ __launch_bounds__(256) void k_flow_warp(
    const float* __restrict__ warp_input, const float* __restrict__ pred,
    const float* __restrict__ diff, const float* __restrict__ wv,
    float* __restrict__ img_out, float* __restrict__ flow_out) {
  const int b     = blockIdx.x >> 8;
  const int pbase = (blockIdx.x & 255) << 8;
  const int t     = threadIdx.x;
  __shared__ float sc[32];  // dst control points, (y,x) interleaved
  __shared__ float sw[38];  // spline solution rows 0..18 x {y,x}
  if (t < 32)       sc[t]      = pred[b * 32 + t] + diff[b * 32 + t];
  else if (t < 70)  sw[t - 32] = wv[b * 38 + (t - 32)];
  __syncthreads();

  const int   p  = pbase + t;
  const float fy = (float)(p >> 8);
  const float fx = (float)(p & 255);

  // affine part: q_aug @ v   (v rows are sol[16..18])
  float f0 = fy * sw[32] + fx * sw[34] + sw[36];
  float f1 = fy * sw[33] + fx * sw[35] + sw[37];
#pragma unroll
  for (int i = 0; i < 16; ++i) {
    const float dy = fy - sc[2 * i], dx = fx - sc[2 * i + 1];
    const float r2 = dy * dy + dx * dx;
    const float ph = 0.5f * r2 * __logf(fmaxf(r2, 1e-10f));
    f0 = fmaf(ph, sw[2 * i], f0);
    f1 = fmaf(ph, sw[2 * i + 1], f1);
  }

  const size_t pix = (size_t)b * HW + p;
  flow_out[pix * 2]     = f0;
  flow_out[pix * 2 + 1] = f1;

  // bilinear warp at coords = grid - flow
  float cy = fminf(fmaxf(fy - f0, 0.0f), (float)(HH - 1));
  float cx = fminf(fmaxf(fx - f1, 0.0f), (float)(WW - 1));
  const int y0 = (int)fminf(floorf(cy), (float)(HH - 2));
  const int x0 = (int)fminf(floorf(cx), (float)(WW - 2));
  const float ay = cy - (float)y0, ax = cx - (float)x0;
  const float w00 = (1.0f - ay) * (1.0f - ax);
  const float w01 = (1.0f - ay) * ax;
  const float w10 = ay * (1.0f - ax);
  const float w11 = ay * ax;
  const float* im = warp_input + (size_t)b * HW * 3 + (size_t)(y0 * WW + x0) * 3;
#pragma unroll
  for (int c = 0; c < 3; ++c) {
    const float o = w00 * im[c] + w01 * im[3 + c] +
                    w10 * im[WW * 3 + c] + w11 * im[WW * 3 + 3 + c];
    img_out[pix * 3 + c] = o;
  }
}

// --------------------------------------------------------------------------
extern "C" void kernel_launch(void* const* d_in, const int* in_sizes, int n_in,
                              void* d_out, int out_size, void* d_ws, size_t ws_size,
                              hipStream_t stream) {
  const float* x       = (const float*)d_in[0];
  const float* warp_in = (const float*)d_in[1];
  const float* scales  = (const float*)d_in[2];
  const float* W1      = (const float*)d_in[3];
  const float* b1      = (const float*)d_in[4];
  const float* gamma   = (const float*)d_in[5];
  const float* beta    = (const float*)d_in[6];
  const float* W2      = (const float*)d_in[7];
  const float* W3      = (const float*)d_in[8];
  const float* lmean   = (const float*)d_in[9];

  float* out      = (float*)d_out;
  float* img_out  = out;                                // B*H*W*3
  float* flow_out = out + (size_t)B_N * HW * 3;         // B*H*W*2
  float* pred_out = flow_out + (size_t)B_N * HW * 2;    // B*32
  float* diff_out = pred_out + B_N * 32;                // B*32

  float* h_raw = (float*)d_ws;          // B*128
  float* wv    = h_raw + B_N * HID;     // B*38 (spline solution rows 0..18 x 2)

  k_gemm_h<<<dim3(16), dim3(32), 0, stream>>>(x, W1, h_raw);
  k_ln_ldmark<<<dim3(B_N), dim3(128), 0, stream>>>(h_raw, b1, gamma, beta, W2, W3,
                                                   lmean, scales, pred_out, diff_out);
  k_spline_solve<<<dim3(B_N), dim3(32), 0, stream>>>(pred_out, diff_out, wv);
  k_flow_warp<<<dim3(B_N * 256), dim3(256), 0, stream>>>(warp_in, pred_out, diff_out,
                                                         wv, img_out, flow_out);
}